// LogitSeparator_30992484008164
// MI455X (gfx1250) — compile-verified
//
#include <hip/hip_runtime.h>
#include <hip/hip_bf16.h>
#include <stdint.h>

typedef __attribute__((ext_vector_type(4))) float v4f;

// One block per (b, d) output row. blockIdx.x = b*32 + d.
__global__ __launch_bounds__(256) void LogitSeparator_kernel(
    const int* __restrict__ schemas,   // (B, 32) int32
    const float* __restrict__ logits,  // (B, L)  float32
    float* __restrict__ out,           // (B, 32, L)
    float* __restrict__ omask,         // (B, 32, L)
    int L) {
  constexpr int D = 32;
  const int bd = blockIdx.x;
  const int b = bd / D;
  const int d = bd % D;
  const int tid = (int)threadIdx.x;

  __shared__ int s_start, s_len;
  __shared__ float zone[256];  // max schema value is 255

  // --- wave32 exclusive-scan of schemas[b, :] (D == warpSize == 32) ---
  if (tid < 32) {
    int x = schemas[b * D + tid];
    int incl = x;
#pragma unroll
    for (int off = 1; off < 32; off <<= 1) {
      int y = __shfl_up(incl, off, 32);
      if (tid >= off) incl += y;
    }
    if (tid == d) {
      s_start = incl - x;  // exclusive prefix sum
      s_len = x;
    }
  }
  __syncthreads();

  // Wave-uniform by construction; make them scalar for SGPR addressing.
  const int start = __builtin_amdgcn_readfirstlane(s_start);
  const int len   = __builtin_amdgcn_readfirstlane(s_len);

  // --- CDNA5 async global->LDS staging of the zone (ASYNCcnt path) ---
  // Uniform 64-bit base in SGPR pair, per-lane 32-bit byte offset in VGPR.
  const unsigned long long src_base =
      (unsigned long long)(uintptr_t)(logits + (size_t)b * L + start);
  if (tid < len) {
    unsigned lds_off = (unsigned)(uintptr_t)(&zone[0]) + (unsigned)tid * 4u;
    unsigned goff = (unsigned)tid * 4u;
    asm volatile("global_load_async_to_lds_b32 %0, %1, %2"
                 :
                 : "v"(lds_off), "v"(goff), "s"(src_base)
                 : "memory");
  }
  asm volatile("s_wait_asynccnt 0" ::: "memory");
  __syncthreads();

  float* __restrict__ orow = out + (size_t)bd * L;
  float* __restrict__ mrow = omask + (size_t)bd * L;
  const v4f zero = {0.0f, 0.0f, 0.0f, 0.0f};

  // Chunk 0: k = tid*4 spans [0, 1024) and fully contains the zone (len<=255).
  {
    const int k = tid << 2;
    v4f val = zero;
    v4f m = zero;
    if (k < len) {
#pragma unroll
      for (int j = 0; j < 4; ++j) {
        if (k + j < len) {
          val[j] = zone[k + j];
          m[j] = 1.0f;
        }
      }
    }
    __builtin_nontemporal_store(val, (v4f*)(orow + k));
    __builtin_nontemporal_store(m, (v4f*)(mrow + k));
  }

  // Chunks 1..7: branchless non-temporal zero fill (pure b128 store stream).
  for (int k = 1024 + (tid << 2); k < L; k += 1024) {
    __builtin_nontemporal_store(zero, (v4f*)(orow + k));
    __builtin_nontemporal_store(zero, (v4f*)(mrow + k));
  }
}

extern "C" void kernel_launch(void* const* d_in, const int* in_sizes, int n_in,
                              void* d_out, int out_size, void* d_ws, size_t ws_size,
                              hipStream_t stream) {
  const int* schemas = (const int*)d_in[0];    // (B, 32) integer input
  const float* logits = (const float*)d_in[1]; // (B, L) float32

  const int n_rows = in_sizes[0];  // B * D = B * 32
  const int B = n_rows / 32;
  const int L = in_sizes[1] / B;   // 8192

  float* out = (float*)d_out;               // first output: B*32*L floats
  float* omask = out + (size_t)n_rows * L;  // second output: mask as 0/1 floats

  dim3 grid(n_rows);
  dim3 block(256);
  hipLaunchKernelGGL(LogitSeparator_kernel, grid, block, 0, stream,
                     schemas, logits, out, omask, L);
}